// Net_16982300688664
// MI455X (gfx1250) — compile-verified
//
#include <hip/hip_runtime.h>
#include <hip/hip_bf16.h>
#include <math.h>

// ---------------------------------------------------------------------------
// GNN-with-edge-noise forward:
//   per layer: a = a_mu + exp(a_log_sigma)*eps ; m = a * h[src] ;
//              agg = segment_sum(m, dst) ; out = relu?(agg @ W + b)
//   nll = mean((a0-1)^2) + log(2*pi)     (prior sigma == 1)
// ---------------------------------------------------------------------------

typedef __attribute__((ext_vector_type(2))) float v2f;
typedef __attribute__((ext_vector_type(4))) float v4f;
typedef __attribute__((ext_vector_type(8))) float v8f;

#define FEAT 32          // in/hid feature width (all three gathers are 32-wide)
#define CHUNKS 8         // 32 feats / 4-wide vector

__device__ inline void atomAddF32(float* p, float v) {
    __hip_atomic_fetch_add(p, v, __ATOMIC_RELAXED, __HIP_MEMORY_SCOPE_AGENT);
}

// --------------------------- zero fill -------------------------------------
__global__ void zero_kernel(float* __restrict__ p, size_t n) {
    size_t i = (size_t)blockIdx.x * blockDim.x + threadIdx.x;
    size_t stride = (size_t)gridDim.x * blockDim.x;
    for (; i < n; i += stride) p[i] = 0.0f;
}

// --------------------------- edge phase ------------------------------------
// One thread handles 4 features of one edge (8 threads / edge).
// eps is pure streaming -> non-temporal loads. h[src] table is 12.8MB -> L2
// resident gather. agg is 12.8MB -> L2-resident f32 atomics.
template <bool NLL>
__global__ __launch_bounds__(256) void edge_kernel(
    const float* __restrict__ h, const float* __restrict__ eps,
    const int* __restrict__ src, const int* __restrict__ dst,
    const float* __restrict__ a_mu_p, const float* __restrict__ a_ls_p,
    float* __restrict__ agg, float* __restrict__ nll_accum, int n_edges) {
    __shared__ float red[256];
    const int idx = blockIdx.x * 256 + threadIdx.x;
    float local = 0.0f;
    if (idx < n_edges * CHUNKS) {
        const int e = idx >> 3;
        const int c = idx & 7;
        const float amu = a_mu_p[0];
        const float sg  = expf(a_ls_p[0]);

        const v4f* epsv = (const v4f*)(eps + (size_t)e * FEAT + c * 4);
        v4f ev = __builtin_nontemporal_load(epsv);         // streaming input
        const int s = src[e];
        const int d = dst[e];
        v4f hv = *(const v4f*)(h + (size_t)s * FEAT + c * 4); // L2-hit gather

        v4f a;
        a.x = fmaf(sg, ev.x, amu);
        a.y = fmaf(sg, ev.y, amu);
        a.z = fmaf(sg, ev.z, amu);
        a.w = fmaf(sg, ev.w, amu);

        float* ap = agg + (size_t)d * FEAT + c * 4;
        atomAddF32(ap + 0, a.x * hv.x);
        atomAddF32(ap + 1, a.y * hv.y);
        atomAddF32(ap + 2, a.z * hv.z);
        atomAddF32(ap + 3, a.w * hv.w);

        if (NLL) {
            float dx = a.x - 1.0f, dy = a.y - 1.0f, dz = a.z - 1.0f, dw = a.w - 1.0f;
            local = dx * dx + dy * dy + dz * dz + dw * dw;
        }
    }
    if (NLL) {
        red[threadIdx.x] = local;
        __syncthreads();
        for (int off = 128; off > 0; off >>= 1) {
            if ((int)threadIdx.x < off) red[threadIdx.x] += red[threadIdx.x + off];
            __syncthreads();
        }
        if (threadIdx.x == 0) atomAddF32(nll_accum, red[0]);
    }
}

// --------------------------- WMMA fp32 GEMM --------------------------------
// out[M x OF] = relu?(A[M x KF] @ W[KF x OF] + bias), fp32 throughout via
// V_WMMA_F32_16X16X4_F32. One wave computes one 16x16 output tile (K = KF as
// KF/4 chained K=4 WMMAs).
//
// f32 16x4 A layout: lanes 0-15 -> M=lane, {vgpr0,vgpr1} = K=0,1;
//                    lanes 16-31 -> M=lane-16, {vgpr0,vgpr1} = K=2,3.
// f32 16x16 C/D layout: vgpr v, lanes 0-15 -> (M=v, N=lane);
//                       lanes 16-31 -> (M=v+8, N=lane-16).
template <int KF, int OF, bool RELU>
__global__ __launch_bounds__(128) void gemm_bias_kernel(
    const float* __restrict__ A, const float* __restrict__ W,
    const float* __restrict__ bias, float* __restrict__ out, int tiles_m) {
    const int tiles_n = OF / 16;
    const int wave = blockIdx.x * (blockDim.x >> 5) + (threadIdx.x >> 5);
    const int lane = threadIdx.x & 31;
    const int tm = wave / tiles_n;
    const int tn = wave % tiles_n;
    if (tm >= tiles_m) return;   // uniform per-wave exit: EXEC stays all-ones

    const int hi  = lane >> 4;   // 0: K pair {0,1}, 1: K pair {2,3}
    const int l16 = lane & 15;

    const float* arow = A + ((size_t)tm * 16 + l16) * KF;
    const float* wcol = W + (size_t)tn * 16 + l16;

    v8f acc = {};
#pragma unroll
    for (int s = 0; s < KF / 4; ++s) {
        const int k0 = s * 4 + hi * 2;
        v2f av, bv;
        av.x = arow[k0];
        av.y = arow[k0 + 1];
        bv.x = wcol[(size_t)k0 * OF];
        bv.y = wcol[(size_t)(k0 + 1) * OF];
        acc = __builtin_amdgcn_wmma_f32_16x16x4_f32(
            /*neg_a=*/false, av, /*neg_b=*/false, bv,
            /*c_mod=*/(short)0, acc, /*reuse_a=*/false, /*reuse_b=*/false);
    }

    const float bval = bias[tn * 16 + l16];
#pragma unroll
    for (int v = 0; v < 8; ++v) {
        float val = acc[v] + bval;
        if (RELU) val = fmaxf(val, 0.0f);
        out[((size_t)tm * 16 + v + hi * 8) * OF + tn * 16 + l16] = val;
    }
}

// --------------------------- finalize scalar -------------------------------
__global__ void finalize_kernel(const float* __restrict__ nll_sum,
                                float* __restrict__ out_scalar, float inv_count) {
    // -2 * mean(logprob) = mean((a-1)^2) + log(2*pi)   (sigma_prior == 1)
    out_scalar[0] = nll_sum[0] * inv_count + 1.83787706640934548356f;
}

// ---------------------------------------------------------------------------
extern "C" void kernel_launch(void* const* d_in, const int* in_sizes, int n_in,
                              void* d_out, int out_size, void* d_ws, size_t ws_size,
                              hipStream_t stream) {
    const float* x    = (const float*)d_in[0];
    const float* W0   = (const float*)d_in[1];
    const float* b0   = (const float*)d_in[2];
    const float* W1   = (const float*)d_in[3];
    const float* b1   = (const float*)d_in[4];
    const float* W2   = (const float*)d_in[5];
    const float* b2   = (const float*)d_in[6];
    const float* a_mu = (const float*)d_in[7];
    const float* a_ls = (const float*)d_in[8];
    const float* eps0 = (const float*)d_in[9];
    const float* eps1 = (const float*)d_in[10];
    const float* eps2 = (const float*)d_in[11];
    const int*   src  = (const int*)d_in[12];
    const int*   dst  = (const int*)d_in[13];

    const int n_nodes = in_sizes[0] / FEAT;       // 100000
    const int n_edges = in_sizes[9] / FEAT;       // 1600000
    const size_t nodeF = (size_t)n_nodes * FEAT;  // 3.2M floats

    float* agg  = (float*)d_ws;                   // [N,32]
    float* hA   = agg + nodeF;                    // [N,32]
    float* hB   = hA + nodeF;                     // [N,32]
    float* nll  = hB + nodeF;                     // [1]

    float* out_h   = (float*)d_out;               // [N,16]
    float* out_nll = out_h + (size_t)n_nodes * 16;

    const int zeroBlocks = 2048;
    const int edgeBlocks = (n_edges * CHUNKS + 255) / 256;
    const int tiles_m = (n_nodes + 15) / 16;      // 6250 (divides exactly)
    const int gemmBlocks32 = (tiles_m * 2 + 3) / 4;  // OF=32 -> 2 n-tiles
    const int gemmBlocks16 = (tiles_m * 1 + 3) / 4;  // OF=16 -> 1 n-tile

    // ---- layer 0: x -> hA (relu), plus NLL accumulation --------------------
    zero_kernel<<<zeroBlocks, 256, 0, stream>>>(agg, nodeF);
    zero_kernel<<<1, 1, 0, stream>>>(nll, 1);
    edge_kernel<true><<<edgeBlocks, 256, 0, stream>>>(
        x, eps0, src, dst, a_mu, a_ls, agg, nll, n_edges);
    gemm_bias_kernel<32, 32, true><<<gemmBlocks32, 128, 0, stream>>>(
        agg, W0, b0, hA, tiles_m);

    // ---- layer 1: hA -> hB (relu) -----------------------------------------
    zero_kernel<<<zeroBlocks, 256, 0, stream>>>(agg, nodeF);
    edge_kernel<false><<<edgeBlocks, 256, 0, stream>>>(
        hA, eps1, src, dst, a_mu, a_ls, agg, nullptr, n_edges);
    gemm_bias_kernel<32, 32, true><<<gemmBlocks32, 128, 0, stream>>>(
        agg, W1, b1, hB, tiles_m);

    // ---- layer 2: hB -> d_out (no relu) -----------------------------------
    zero_kernel<<<zeroBlocks, 256, 0, stream>>>(agg, nodeF);
    edge_kernel<false><<<edgeBlocks, 256, 0, stream>>>(
        hB, eps2, src, dst, a_mu, a_ls, agg, nullptr, n_edges);
    gemm_bias_kernel<32, 16, false><<<gemmBlocks16, 128, 0, stream>>>(
        agg, W2, b2, out_h, tiles_m);

    // ---- scalar regularizer ------------------------------------------------
    finalize_kernel<<<1, 1, 0, stream>>>(
        nll, out_nll, 1.0f / ((float)n_edges * (float)FEAT));
}